// InfoNCELoss3_34110630265479
// MI455X (gfx1250) — compile-verified
//
#include <hip/hip_runtime.h>
#include <stdint.h>

// ---------------------------------------------------------------------------
// InfoNCE loss for MI455X (gfx1250).  B*N = total nodes, D = 128 (f32).
// Stable counting sort by group, positive = prev-in-group (wrap), negative =
// threefry2x32(key=(0,42)) uniform over complement (bit-exact jax reproduction),
// f32 dot products via V_WMMA_F32_16X16X4_F32 (diagonal of 16x16 Gram tiles).
// ---------------------------------------------------------------------------

#define G_NUM  1024
#define GP1    1025          // groups + masked sentinel
#define EPB    1024          // elements per sort block
#define D_DIM  128

typedef __attribute__((ext_vector_type(2))) float v2f;
typedef __attribute__((ext_vector_type(8))) float v8f;

__device__ __forceinline__ unsigned rotl32(unsigned x, int n) {
  return (x << n) | (x >> (32 - n));
}
__device__ __forceinline__ void tf_mix(unsigned& x0, unsigned& x1, int r) {
  x0 += x1; x1 = rotl32(x1, r); x1 ^= x0;
}

// jax threefry2x32 with key (0, 42); counter layout of jax.random.uniform over
// a flat array of `total` elements: out[i] = x0'(j) if i<half else x1'(j), j=i%half.
__device__ __forceinline__ unsigned threefry_jax(unsigned i, unsigned half) {
  unsigned j  = (i < half) ? i : i - half;
  unsigned x0 = j, x1 = j + half;
  const unsigned ks0 = 0u, ks1 = 42u;
  const unsigned ks2 = 0x1BD11BDAu ^ ks0 ^ ks1;
  x0 += ks0; x1 += ks1;
  tf_mix(x0, x1, 13); tf_mix(x0, x1, 15); tf_mix(x0, x1, 26); tf_mix(x0, x1, 6);
  x0 += ks1; x1 += ks2 + 1u;
  tf_mix(x0, x1, 17); tf_mix(x0, x1, 29); tf_mix(x0, x1, 16); tf_mix(x0, x1, 24);
  x0 += ks2; x1 += ks0 + 2u;
  tf_mix(x0, x1, 13); tf_mix(x0, x1, 15); tf_mix(x0, x1, 26); tf_mix(x0, x1, 6);
  x0 += ks0; x1 += ks1 + 3u;
  tf_mix(x0, x1, 17); tf_mix(x0, x1, 29); tf_mix(x0, x1, 16); tf_mix(x0, x1, 24);
  x0 += ks1; x1 += ks2 + 4u;
  tf_mix(x0, x1, 13); tf_mix(x0, x1, 15); tf_mix(x0, x1, 26); tf_mix(x0, x1, 6);
  x0 += ks2; x1 += ks0 + 5u;
  return (i < half) ? x0 : x1;
}

// --- K1: per-block histogram over GP1 buckets + g_eff materialization -------
__global__ void k_hist(const int* __restrict__ groups,
                       const unsigned char* __restrict__ mask,
                       int* __restrict__ geff, int* __restrict__ hist) {
  __shared__ int h[GP1];
  for (int i = threadIdx.x; i < GP1; i += blockDim.x) h[i] = 0;
  __syncthreads();
  const int base = blockIdx.x * EPB;
  for (int j = threadIdx.x; j < EPB; j += blockDim.x) {
    const int e = base + j;
    const int g = (mask[e] != 0) ? groups[e] : G_NUM;
    geff[e] = g;
    atomicAdd(&h[g], 1);
  }
  __syncthreads();
  int* out = hist + (size_t)blockIdx.x * GP1;
  for (int i = threadIdx.x; i < GP1; i += blockDim.x) out[i] = h[i];
}

// --- K2: per-group exclusive scan across sort-blocks ------------------------
__global__ void k_scan_blocks(int* __restrict__ hist, int* __restrict__ counts,
                              int numBlocks) {
  const int g = blockIdx.x;
  const int t = threadIdx.x;
  const int lane = t & 31, wid = t >> 5;
  int v = (t < numBlocks) ? hist[(size_t)t * GP1 + g] : 0;
  int x = v;
  #pragma unroll
  for (int o = 1; o < 32; o <<= 1) { int y = __shfl_up(x, o); if (lane >= o) x += y; }
  __shared__ int wsum[8];
  if (lane == 31) wsum[wid] = x;
  __syncthreads();
  if (t < 8) {
    int y = wsum[t];
    #pragma unroll
    for (int o = 1; o < 8; o <<= 1) { int z = __shfl_up(y, o); if (t >= o) y += z; }
    wsum[t] = y;
  }
  __syncthreads();
  const int base = wid ? wsum[wid - 1] : 0;
  const int incl = x + base;
  if (t < numBlocks) hist[(size_t)t * GP1 + g] = incl - v;   // exclusive offset
  if (t == numBlocks - 1) counts[g] = incl;                  // group total
}

// --- K3: exclusive scan over group counts -> starts, valid_total ------------
__global__ void k_scan_groups(const int* __restrict__ counts,
                              int* __restrict__ starts, int* __restrict__ vt,
                              int total) {
  if (threadIdx.x == 0 && blockIdx.x == 0) {
    int run = 0;
    for (int g = 0; g < GP1; ++g) { starts[g] = run; run += counts[g]; }
    vt[0] = total - counts[G_NUM];
  }
}

// --- K4: stable scatter (counting sort) -------------------------------------
__global__ void k_scatter(const int* __restrict__ geff,
                          const int* __restrict__ boffAll,
                          const int* __restrict__ starts,
                          int* __restrict__ order, int* __restrict__ sg) {
  __shared__ int lg[EPB];
  const int base = blockIdx.x * EPB;
  for (int j = threadIdx.x; j < EPB; j += blockDim.x) lg[j] = geff[base + j];
  __syncthreads();
  const int* boff = boffAll + (size_t)blockIdx.x * GP1;
  for (int j0 = 0; j0 < EPB; j0 += blockDim.x) {
    const int p = j0 + threadIdx.x;
    const int g = lg[p];
    int rank = 0;
    for (int q = 0; q < p; ++q) rank += (lg[q] == g);  // stable intra-block rank
    const int pos = starts[g] + boff[g] + rank;
    order[pos] = base + p;
    sg[pos] = g;
  }
}

// --- K5: WMMA loss over 16-position tiles (one wave per tile) ---------------
__global__ void k_loss(const float* __restrict__ emb,
                       const int* __restrict__ order,
                       const int* __restrict__ sg,
                       const int* __restrict__ counts,
                       const int* __restrict__ starts,
                       const int* __restrict__ vtp,
                       float* __restrict__ partials, int total) {
  const int lane = threadIdx.x & 31;
  const int wid  = threadIdx.x >> 5;
  __shared__ float accL[8], accW[8];

  const int tile = blockIdx.x * 8 + wid;
  const int i0   = tile * 16;
  const int vt   = vtp[0];

  float cl = 0.f, cw = 0.f;
  if (i0 < vt) {   // whole-tile skip for masked (sentinel) tail: wave-uniform
    const unsigned half = (unsigned)(total >> 1);
    // Lanes 0..15 own m = lane; lanes 16..31 redundantly mirror m = lane-16.
    const int m = lane & 15;
    const int i = i0 + m;
    const int g   = sg[i];
    const int st  = starts[g];
    const int cnt = counts[g];
    const int cnt1 = (cnt > 1) ? cnt : 1;
    const int local = i - st;
    int pl = (local - 1) % cnt1; if (pl < 0) pl += cnt1;   // jnp.mod semantics
    const int aidx = order[i];
    const int pidx = order[st + pl];
    const unsigned bits = threefry_jax((unsigned)i, half);
    const float u = __uint_as_float((bits >> 9) | 0x3f800000u) - 1.0f;
    int comp = vt - cnt; if (comp < 1) comp = 1;
    int r = (int)(u * (float)comp);
    if (r > comp - 1) r = comp - 1;
    int np = (r < st) ? r : r + cnt;
    if (np < 0) np = 0; if (np > total - 1) np = total - 1;
    const int nidx = order[np];
    const bool valid = (g < G_NUM) && (cnt >= 2) && ((vt - cnt) > 0);
    const float wv = valid ? 1.f : 0.f;

    const float* ar = emb + (size_t)aidx * D_DIM;
    const float* pr = emb + (size_t)pidx * D_DIM;
    const float* nr = emb + (size_t)nidx * D_DIM;
    const int ko = (lane >> 4) << 1;   // lanes 0..15: K={0,1}; 16..31: K={2,3}

    v8f cp = {}; v8f cn = {};
    #pragma unroll 4
    for (int c = 0; c < D_DIM / 4; ++c) {
      const int off = (c << 2) + ko;
      v2f av = *(const v2f*)(ar + off);
      v2f pv = *(const v2f*)(pr + off);
      v2f nv = *(const v2f*)(nr + off);
      cp = __builtin_amdgcn_wmma_f32_16x16x4_f32(false, av, false, pv,
                                                 (short)0, cp, false, false);
      cn = __builtin_amdgcn_wmma_f32_16x16x4_f32(false, av, false, nv,
                                                 (short)0, cn, false, false);
    }
    // Diagonal C[m][m]: lanes 0..7 (m=lane, VGPR m) and 24..31 (m=lane-16, VGPR m-8).
    const bool has = (lane < 8) || (lane >= 24);
    const int sel = lane & 7;
    float ps = cp[0], ns = cn[0];
    #pragma unroll
    for (int k = 1; k < 8; ++k) { if (sel == k) { ps = cp[k]; ns = cn[k]; } }
    ps *= 10.0f;  // 1/TEMPERATURE
    ns *= 10.0f;
    const float dd = ns - ps;
    const float li = (dd > 0.f) ? (dd + log1pf(expf(-dd))) : log1pf(expf(dd));
    cl = has ? li * wv : 0.f;
    cw = has ? wv : 0.f;
  }
  #pragma unroll
  for (int o = 16; o > 0; o >>= 1) { cl += __shfl_down(cl, o); cw += __shfl_down(cw, o); }
  if (lane == 0) { accL[wid] = cl; accW[wid] = cw; }
  __syncthreads();
  if (threadIdx.x == 0) {
    float sl = 0.f, sw = 0.f;
    for (int k = 0; k < 8; ++k) { sl += accL[k]; sw += accW[k]; }
    partials[2 * blockIdx.x]     = sl;
    partials[2 * blockIdx.x + 1] = sw;
  }
}

// --- K6: final reduction ----------------------------------------------------
__global__ void k_final(const float* __restrict__ partials, int nPart,
                        float* __restrict__ out) {
  __shared__ float sL[256], sW[256];
  float l = 0.f, w = 0.f;
  for (int i = threadIdx.x; i < nPart; i += blockDim.x) {
    l += partials[2 * i]; w += partials[2 * i + 1];
  }
  sL[threadIdx.x] = l; sW[threadIdx.x] = w;
  __syncthreads();
  for (int s = 128; s > 0; s >>= 1) {
    if (threadIdx.x < s) { sL[threadIdx.x] += sL[threadIdx.x + s];
                           sW[threadIdx.x] += sW[threadIdx.x + s]; }
    __syncthreads();
  }
  if (threadIdx.x == 0) out[0] = sL[0] / fmaxf(sW[0], 1.0f);
}

extern "C" void kernel_launch(void* const* d_in, const int* in_sizes, int n_in,
                              void* d_out, int out_size, void* d_ws, size_t ws_size,
                              hipStream_t stream) {
  const float*         emb    = (const float*)d_in[0];
  const int*           groups = (const int*)d_in[1];
  const unsigned char* mask   = (const unsigned char*)d_in[2];  // jax bool = 1 byte
  float* out = (float*)d_out;

  const int total     = in_sizes[1];        // B*N  (262144)
  const int numBlocks = total / EPB;        // 256
  const int tiles     = total / 16;         // 16384
  const int lossBlks  = tiles / 8;          // 2048

  char* ws = (char*)d_ws;
  size_t off = 0;
  int* geff   = (int*)(ws + off); off += (size_t)total * 4;
  int* hist   = (int*)(ws + off); off += (size_t)numBlocks * GP1 * 4;
  int* counts = (int*)(ws + off); off += (size_t)GP1 * 4;
  int* starts = (int*)(ws + off); off += (size_t)GP1 * 4;
  int* order  = (int*)(ws + off); off += (size_t)total * 4;
  int* sg     = (int*)(ws + off); off += (size_t)total * 4;
  int* vt     = (int*)(ws + off); off += 64;
  float* partials = (float*)(ws + off); off += (size_t)lossBlks * 2 * 4;

  hipLaunchKernelGGL(k_hist,        dim3(numBlocks), dim3(256), 0, stream,
                     groups, mask, geff, hist);
  hipLaunchKernelGGL(k_scan_blocks, dim3(GP1),       dim3(256), 0, stream,
                     hist, counts, numBlocks);
  hipLaunchKernelGGL(k_scan_groups, dim3(1),         dim3(32),  0, stream,
                     counts, starts, vt, total);
  hipLaunchKernelGGL(k_scatter,     dim3(numBlocks), dim3(256), 0, stream,
                     geff, hist, starts, order, sg);
  hipLaunchKernelGGL(k_loss,        dim3(lossBlks),  dim3(256), 0, stream,
                     emb, order, sg, counts, starts, vt, partials, total);
  hipLaunchKernelGGL(k_final,       dim3(1),         dim3(256), 0, stream,
                     partials, lossBlks, out);
}